// SolverNet_9938554323601
// MI455X (gfx1250) — compile-verified
//
#include <hip/hip_runtime.h>

#define NNODE       1024
#define NDIM        2048     // N
#define ROWSTRIDE   2049     // N+1 (stiff columns + resid column)
#define NWG         64
#define BLOCK       256      // 8 wave32 waves
#define ROWS_PER_WG 32       // NDIM / NWG
#define NITER       2048
#define LDS_STRIDE  2052     // 2048 + 4 float pad -> 4-bank skew per row, conflict-free b64 reads

// dynamic LDS layout (floats): A stripe | p vector | per-wave partials
#define LDS_A_FLOATS   (ROWS_PER_WG * LDS_STRIDE)          // 65664
#define LDS_P_OFF      LDS_A_FLOATS
#define LDS_PART_OFF   (LDS_A_FLOATS + NDIM)
#define SMEM_FLOATS    (LDS_A_FLOATS + NDIM + 128)
#define SMEM_BYTES     (SMEM_FLOATS * 4)                   // 271,360 B < 320 KB WGP LDS

typedef float v2f __attribute__((ext_vector_type(2)));
typedef float v8f __attribute__((ext_vector_type(8)));

__device__ __forceinline__ float wave_sum32(float v) {
#pragma unroll
    for (int off = 16; off > 0; off >>= 1)
        v += __shfl_xor(v, off, 32);
    return v;
}

// Sense-free monotonic grid barrier: counter only increments; target = #barriers * NWG.
__device__ __forceinline__ void grid_barrier(unsigned* cnt, unsigned target) {
    __builtin_amdgcn_fence(__ATOMIC_RELEASE, "agent");   // flush this thread's stores to device scope
    __syncthreads();
    if (threadIdx.x == 0) {
        __hip_atomic_fetch_add(cnt, 1u, __ATOMIC_RELAXED, __HIP_MEMORY_SCOPE_AGENT);
        while (__hip_atomic_load(cnt, __ATOMIC_RELAXED, __HIP_MEMORY_SCOPE_AGENT) < target)
            __builtin_amdgcn_s_sleep(1);
    }
    __syncthreads();
    __builtin_amdgcn_fence(__ATOMIC_ACQUIRE, "agent");   // invalidate WGP$ so we see remote writes
}

__global__ void cg_init(unsigned* cnt) { *cnt = 0u; }

__global__ __launch_bounds__(BLOCK, 1)
void cg_solver(const float* __restrict__ x,
               const float* __restrict__ xcoord,
               const float* __restrict__ ycoord,
               float* __restrict__ out,
               float* __restrict__ pvec,      // [NDIM] shared search direction
               float* __restrict__ part_a,    // [NWG] p.Ap partials
               float* __restrict__ part_b,    // [NWG] r.r partials
               unsigned* cnt) {
    extern __shared__ float smem[];
    float* ldsA     = smem;                   // [32][LDS_STRIDE] matrix stripe
    float* ldsP     = smem + LDS_P_OFF;       // [2048] p staged per iteration
    float* lds_part = smem + LDS_PART_OFF;    // [8][16] per-wave Ap partials

    const int wg   = blockIdx.x;
    const int tid  = threadIdx.x;
    const int wave = tid >> 5;
    const int lane = tid & 31;

    // ---- one-time async stage of this WG's 32-row stripe into LDS (CDNA5 async-to-LDS) ----
    // 65536 dwords / 256 threads = 256 async b32 per thread; hardware throttles ASYNCcnt.
    {
        const int rbase0 = wg * ROWS_PER_WG;
#pragma unroll 4
        for (int i = 0; i < (ROWS_PER_WG * NDIM) / BLOCK; ++i) {
            const int idx  = i * BLOCK + tid;
            const int row  = idx >> 11;                 // /2048
            const int k    = idx & (NDIM - 1);
            unsigned loff  = (unsigned)(size_t)&ldsA[row * LDS_STRIDE + k];
            unsigned long long gaddr =
                (unsigned long long)(size_t)(x + (size_t)(rbase0 + row) * ROWSTRIDE + k);
            asm volatile("global_load_async_to_lds_b32 %0, %1, off"
                         :: "v"(loff), "v"(gaddr) : "memory");
        }
    }

    // WMMA tiling: wave -> (stripe s of 16 rows, K-chunk q of 512)
    const int s   = wave >> 2;
    const int q   = wave & 3;
    const int m   = lane & 15;
    const int hi  = lane >> 4;                // lanes 16-31 hold K+2,K+3 (ISA 16x4 f32 A layout)
    const int kk0 = q * 512 + hi * 2;
    const float* Alds = ldsA + (s * 16 + m) * LDS_STRIDE + kk0;
    const float* Plds = ldsP + kk0;

    // Wave-0 threads own one row each of the CG vectors in registers.
    const int  row = wg * ROWS_PER_WG + tid;            // valid for tid < 32
    float u = 0.0f, r = 0.0f, p_reg = 0.0f, rr = 0.0f;

    unsigned bar = 0;

    // ---- init: r0 = resid, p0 = r0, publish p, partial rr ----
    if (tid < 32) {
        r     = x[(size_t)row * ROWSTRIDE + NDIM];      // resid column
        p_reg = r;
        pvec[row] = p_reg;
        float s0 = wave_sum32(r * r);
        if (lane == 0) part_b[wg] = s0;
    }
    bar += NWG; grid_barrier(cnt, bar);
    if (tid < 32) {
        rr = 0.0f;
#pragma unroll 8
        for (int j = 0; j < NWG; ++j) rr += part_b[j];
    }

    // drain the async stripe copy, make it visible to all waves
    asm volatile("s_wait_asynccnt 0x0" ::: "memory");
    __syncthreads();

    for (int it = 0; it < NITER; ++it) {
        // ---- stage p into LDS (two float4 per thread) ----
        {
            const float4* pv4 = (const float4*)pvec;
            float4*       lp4 = (float4*)ldsP;
            lp4[tid]         = pv4[tid];
            lp4[tid + BLOCK] = pv4[tid + BLOCK];
        }
        __syncthreads();

        // ================= Phase A: Ap = stiff @ p (WMMA from LDS), partial p.Ap =================
        v8f c0 = {0.f,0.f,0.f,0.f,0.f,0.f,0.f,0.f};
        v8f c1 = {0.f,0.f,0.f,0.f,0.f,0.f,0.f,0.f};
#pragma unroll 4
        for (int k = 0; k < 512; k += 8) {
            v2f a0 = *reinterpret_cast<const v2f*>(Alds + k);
            v2f b0 = *reinterpret_cast<const v2f*>(Plds + k);
            c0 = __builtin_amdgcn_wmma_f32_16x16x4_f32(
                     false, a0, false, b0, (short)0, c0, false, false);
            v2f a1 = *reinterpret_cast<const v2f*>(Alds + k + 4);
            v2f b1 = *reinterpret_cast<const v2f*>(Plds + k + 4);
            c1 = __builtin_amdgcn_wmma_f32_16x16x4_f32(
                     false, a1, false, b1, (short)0, c1, false, false);
        }
        // Every column of D is identical; lane 0 carries rows rbase+0..7, lane 16 rows rbase+8..15.
        if (lane == 0) {
#pragma unroll
            for (int i = 0; i < 8; ++i) lds_part[wave * 16 + i] = c0[i] + c1[i];
        } else if (lane == 16) {
#pragma unroll
            for (int i = 0; i < 8; ++i) lds_part[wave * 16 + 8 + i] = c0[i] + c1[i];
        }
        __syncthreads();

        float Ap = 0.0f;
        if (tid < 32) {
            const int s2 = tid >> 4, i2 = tid & 15;     // deterministic fixed-order K-chunk sum
            Ap = lds_part[(s2 * 4 + 0) * 16 + i2] + lds_part[(s2 * 4 + 1) * 16 + i2]
               + lds_part[(s2 * 4 + 2) * 16 + i2] + lds_part[(s2 * 4 + 3) * 16 + i2];
            float pr = wave_sum32(p_reg * Ap);
            if (lane == 0) part_a[wg] = pr;
        }
        bar += NWG; grid_barrier(cnt, bar);

        // ================= Phase B: alpha, update u and r, partial rr_new =================
        if (tid < 32) {
            float pAp = 0.0f;
#pragma unroll 8
            for (int j = 0; j < NWG; ++j) pAp += part_a[j];
            float alpha = rr / pAp;
            u += alpha * p_reg;
            r -= alpha * Ap;
            float s1 = wave_sum32(r * r);
            if (lane == 0) part_b[wg] = s1;
        }
        bar += NWG; grid_barrier(cnt, bar);

        // ================= Phase C: beta, update p, publish p =================
        if (tid < 32) {
            float rrn = 0.0f;
#pragma unroll 8
            for (int j = 0; j < NWG; ++j) rrn += part_b[j];
            float beta = rrn / rr;
            p_reg = r + beta * p_reg;
            pvec[row] = p_reg;
            rr = rrn;
        }
        bar += NWG; grid_barrier(cnt, bar);
    }

    // ---- epilogue: out = [u (2048) | ux (1024) | uy (1024)] ----
    if (tid < 32) {
        out[row] = u;
        const int node = row >> 1;
        if ((row & 1) == 0) out[NDIM + node]         = xcoord[node] + u;
        else                out[NDIM + NNODE + node] = ycoord[node] + u;
    }
}

extern "C" void kernel_launch(void* const* d_in, const int* in_sizes, int n_in,
                              void* d_out, int out_size, void* d_ws, size_t ws_size,
                              hipStream_t stream) {
    (void)in_sizes; (void)n_in; (void)out_size; (void)ws_size;
    const float* x  = (const float*)d_in[0];
    const float* xc = (const float*)d_in[1];
    const float* yc = (const float*)d_in[2];
    float* out = (float*)d_out;

    unsigned* cnt  = (unsigned*)d_ws;          // 128B slot for barrier counter
    float* ws_f    = (float*)d_ws;
    float* part_a  = ws_f + 32;
    float* part_b  = part_a + NWG;
    float* pvec    = part_b + NWG;             // 2048 floats, 16B-aligned from d_ws base

    // Allow >64KB dynamic LDS (320KB per WGP on gfx1250). Host-side, capture-safe, deterministic.
    (void)hipFuncSetAttribute((const void*)cg_solver,
                              hipFuncAttributeMaxDynamicSharedMemorySize, SMEM_BYTES);

    cg_init<<<1, 1, 0, stream>>>(cnt);         // graph-safe counter reset each launch/replay
    cg_solver<<<NWG, BLOCK, SMEM_BYTES, stream>>>(x, xc, yc, out, pvec, part_a, part_b, cnt);
}